// GraphMILNet_27324581937325
// MI455X (gfx1250) — compile-verified
//
#include <hip/hip_runtime.h>
#include <math.h>

// ---------------------------------------------------------------------------
// GraphMIL on MI455X (gfx1250, wave32).
//   features: [8192, 512] f32   W: [512,512] f32   b: [512] f32
//   batch: 64 contiguous bags of 128 rows each; n_topk = 16.
// Outputs (concatenated in d_out, f32):
//   agg: [64, 512]   then  adj: [8192, 8192]  (block-diagonal, 128x128 blocks)
//
// Strategy: hipMemsetAsync zeroes the 256 MB adj (the bandwidth floor of this
// problem), then ONE fused kernel per bag does:
//   1) row norms, 2) fp32 WMMA Gram S = nf nf^T (K-slabbed through LDS,
//   transposed layout so A and B fragments are identical & bank-conflict-free),
//   3) per-row top-16 (>0, lowest-index tie-break), 4) adjacency bitmask,
//   degree, dis=rsqrt(deg), c[j]=dis[j]*sum_i A[j,i]dis[i],
//   5) agg = (c^T @ features_bag) @ W + 128*b   (gcn_out contracted away).
// ---------------------------------------------------------------------------

typedef __attribute__((ext_vector_type(2))) float v2f;
typedef __attribute__((ext_vector_type(8))) float v8f;

#define NBAGS   64
#define BAG     128      // rows per bag
#define D       512      // feature dim
#define NTOPK   16
#define NROWS   8192
#define SLAB    128      // K-slab width staged in LDS
#define ST_A    132      // slabT stride (floats): breaks 64-bank periodicity
#define ST_S    129      // S stride (floats): conflict-free row scans

__global__ __launch_bounds__(256) void graphmil_bag_kernel(
    const float* __restrict__ features,
    const float* __restrict__ W,
    const float* __restrict__ b,
    float* __restrict__ agg_out,   // [64, 512]
    float* __restrict__ adj_out)   // [8192, 8192] (pre-zeroed)
{
    __shared__ float slabT[SLAB * ST_A];     // nf^T slab: slabT[k][m]
    __shared__ float S[BAG * ST_S];          // Gram block
    __shared__ unsigned Abits[BAG * 4];      // adjacency bit rows
    __shared__ float invn[BAG];
    __shared__ float disL[BAG];
    __shared__ float cL[BAG];
    __shared__ float vL[D];

    const int tid  = threadIdx.x;
    const int wv   = tid >> 5;          // wave id 0..7 (wave32)
    const int lane = tid & 31;
    const int half = (lane >> 4) & 1;   // 0: lanes 0-15, 1: lanes 16-31
    const int ln   = lane & 15;
    const int bag  = blockIdx.x;
    const int rowBase = bag * BAG;

    // ---- 1) reciprocal row norms (F.normalize semantics) -------------------
    if (tid < BAG) {
        const float* rp = features + (size_t)(rowBase + tid) * D;
        float s = 0.f;
        for (int k = 0; k < D; ++k) { float x = rp[k]; s += x * x; }
        invn[tid] = 1.0f / fmaxf(sqrtf(s), 1e-12f);
    }
    __syncthreads();

    // ---- 2) Gram S = nf nf^T via V_WMMA_F32_16X16X4_F32 --------------------
    // Wave wv owns output tile-row i = wv (rows 16*wv .. 16*wv+15), all 8
    // column tiles, accumulated across 4 K-slabs of 128.
    v8f acc[8] = {};
    for (int slab = 0; slab < D / SLAB; ++slab) {
        // stage normalized, transposed slab: slabT[k][m] = nf[rowBase+m][kb+k]
        for (int e = tid; e < BAG * SLAB; e += 256) {
            int m = e >> 7, k = e & (SLAB - 1);
            float v = features[(size_t)(rowBase + m) * D + slab * SLAB + k];
            slabT[k * ST_A + m] = v * invn[m];
        }
        __syncthreads();

        for (int k4 = 0; k4 < SLAB; k4 += 4) {
            const int kb = k4 + 2 * half;   // ISA 16x4 f32 A layout
            v2f a;
            a.x = slabT[(kb + 0) * ST_A + 16 * wv + ln];
            a.y = slabT[(kb + 1) * ST_A + 16 * wv + ln];
#pragma unroll
            for (int j = 0; j < 8; ++j) {
                v2f bf;                      // B[k][n] = nf[n0+n][k]
                bf.x = slabT[(kb + 0) * ST_A + 16 * j + ln];
                bf.y = slabT[(kb + 1) * ST_A + 16 * j + ln];
                acc[j] = __builtin_amdgcn_wmma_f32_16x16x4_f32(
                    false, a, false, bf, (short)0, acc[j], false, false);
            }
        }
        __syncthreads();
    }
    // spill accumulators: D VGPR r holds rows {r, r+8} (halves) of the tile
#pragma unroll
    for (int j = 0; j < 8; ++j)
#pragma unroll
        for (int r = 0; r < 8; ++r)
            S[(16 * wv + r + 8 * half) * ST_S + 16 * j + ln] = acc[j][r];
    __syncthreads();

    // ---- 3) per-row top-16 (> 0, lowest-index ties), mask, bitmask ---------
    if (tid < BAG) {
        unsigned keep[4] = {0u, 0u, 0u, 0u};
        float* row = &S[tid * ST_S];
        for (int t = 0; t < NTOPK; ++t) {
            float best = 0.0f;    // strict '>' vs 0: only positives survive,
            int bi = -1;          // matching top_k against out-of-block zeros
            for (int j = 0; j < BAG; ++j) {
                if (!((keep[j >> 5] >> (j & 31)) & 1u)) {
                    float v = row[j];
                    if (v > best) { best = v; bi = j; }
                }
            }
            if (bi < 0) break;
            keep[bi >> 5] |= 1u << (bi & 31);
        }
        // adj output keeps masked similarity values
        for (int j = 0; j < BAG; ++j)
            if (!((keep[j >> 5] >> (j & 31)) & 1u)) row[j] = 0.0f;
        // GCN adjacency: binarized + forced self loop
        keep[tid >> 5] |= 1u << (tid & 31);
        for (int w2 = 0; w2 < 4; ++w2) Abits[tid * 4 + w2] = keep[w2];
    }
    __syncthreads();

    // coalesced write of the masked 128x128 block into adj
    {
        float* blk = adj_out + (size_t)rowBase * NROWS + rowBase;
        for (int e = tid; e < BAG * BAG; e += 256) {
            int r = e >> 7, j = e & (BAG - 1);
            blk[(size_t)r * NROWS + j] = S[r * ST_S + j];
        }
    }

    // ---- 4) dis = rsqrt(colsum(A));  c[j] = dis[j] * sum_i A[j,i] dis[i] ---
    if (tid < BAG) {
        const int i = tid;
        float deg = 0.f;
        for (int j = 0; j < BAG; ++j)
            deg += (float)((Abits[j * 4 + (i >> 5)] >> (i & 31)) & 1u);
        disL[i] = rsqrtf(deg);   // deg >= 1 (self loop)
    }
    __syncthreads();
    if (tid < BAG) {
        const int j = tid;
        float rd = 0.f;
        for (int i = 0; i < BAG; ++i)
            if ((Abits[j * 4 + (i >> 5)] >> (i & 31)) & 1u) rd += disL[i];
        cL[j] = disL[j] * rd;
    }
    __syncthreads();

    // ---- 5) v = c^T @ features_bag ; agg = v @ W + 128*b -------------------
    {
        float v0 = 0.f, v1 = 0.f;
        for (int j = 0; j < BAG; ++j) {
            float cj = cL[j];
            const float* rp = features + (size_t)(rowBase + j) * D;
            v0 += cj * rp[tid];
            v1 += cj * rp[tid + 256];
        }
        vL[tid] = v0;
        vL[tid + 256] = v1;
    }
    __syncthreads();
    {
        float a0 = 0.f, a1 = 0.f;
        for (int d = 0; d < D; ++d) {
            float vd = vL[d];
            a0 += vd * W[(size_t)d * D + tid];
            a1 += vd * W[(size_t)d * D + tid + 256];
        }
        agg_out[(size_t)bag * D + tid]       = a0 + 128.0f * b[tid];
        agg_out[(size_t)bag * D + tid + 256] = a1 + 128.0f * b[tid + 256];
    }
}

extern "C" void kernel_launch(void* const* d_in, const int* in_sizes, int n_in,
                              void* d_out, int out_size, void* d_ws, size_t ws_size,
                              hipStream_t stream) {
    (void)in_sizes; (void)n_in; (void)d_ws; (void)ws_size;
    const float* features = (const float*)d_in[0];
    const float* W        = (const float*)d_in[1];
    const float* b        = (const float*)d_in[2];
    // d_in[3] = batch (structure is contiguous 64x128 by construction)
    // d_in[4] = n_topk (16), d_in[5] = n_bags (64): device scalars, fixed.

    float* out     = (float*)d_out;
    float* agg_out = out;                       // [64*512]
    float* adj_out = out + (size_t)NBAGS * D;   // [8192*8192]

    // Zero entire output (adj is mostly zeros; agg fully overwritten).
    hipMemsetAsync(d_out, 0, (size_t)out_size * sizeof(float), stream);

    graphmil_bag_kernel<<<NBAGS, 256, 0, stream>>>(features, W, b, agg_out, adj_out);
}